// LaplacianLoss_59322088292706
// MI455X (gfx1250) — compile-verified
//
#include <hip/hip_runtime.h>
#include <hip/hip_bf16.h>

// ---------------------------------------------------------------------------
// LaplacianLoss on MI455X (gfx1250):
//   D = L (10000x10000, f32) @ Xw (10000x24, weight folded), then per-batch
//   sum-of-squares -> sqrt -> * NV.  Memory-bound on the 400MB L stream;
//   compute via V_WMMA_F32_16X16X4_F32 so the f32 matrix pipe (not VALU)
//   carries the 12 FLOP/byte intensity.
// ---------------------------------------------------------------------------

#define NV    10000
#define BATCH 8
#define NCOLP 32          // 24 real columns (8 batches x 3 coords), padded to 32
#define TILES (NV / 16)   // 625 row tiles, exact
#define KBLK  64          // K columns of L staged in LDS per step
#define LDS_STRIDE 68     // 64 + 4 pad floats: ds_load_b64 frags hit all 64 banks

typedef __attribute__((ext_vector_type(2))) float v2f;
typedef __attribute__((ext_vector_type(4))) float v4f;
typedef __attribute__((ext_vector_type(8))) float v8f;

// --- Kernel 1: build Xt[c*NV + j] = w * x[b, j, k] (c = 3b+k), cols 24..31 = 0
__global__ void laploss_prep_xt(const float* __restrict__ x,
                                const float* __restrict__ wscalar,
                                float* __restrict__ Xt) {
    int idx = blockIdx.x * blockDim.x + threadIdx.x;
    if (idx >= NCOLP * NV) return;
    int c = idx / NV;
    int j = idx - c * NV;
    float v = 0.0f;
    if (c < BATCH * 3) {
        int b = c / 3;
        int k = c - 3 * b;
        v = wscalar[0] * x[((size_t)b * NV + j) * 3 + k];
    }
    Xt[idx] = v;
}

// --- Kernel 2: one wave (32 lanes) per 16-row tile of L.
// Accumulates two 16x16 f32 WMMA tiles (cols 0-15 and 16-31), then reduces
// the squared outputs deterministically into 8 per-tile batch partials.
__global__ __launch_bounds__(32) void laploss_gemm_tile(
        const float* __restrict__ L,
        const float* __restrict__ Xt,
        float* __restrict__ partial) {
    __shared__ __align__(16) float ldsA[16][LDS_STRIDE];
    __shared__ float red[64];

    const int tile = blockIdx.x;      // 0..624
    const int l    = threadIdx.x;     // 0..31
    const int half = l >> 4;          // WMMA lane half (K split)
    const int m    = l & 15;          // M index (A) / N index (B/C)
    const int r0   = tile * 16;

    v8f c0 = {};   // columns 0..15  (batches 0..5 partially)
    v8f c1 = {};   // columns 16..31 (batches 5..7 + zero pad)

    for (int kb = 0; kb < NV; kb += KBLK) {
        const int kvalid = (NV - kb < KBLK) ? (NV - kb) : KBLK;  // 64 or 16

        // Stage A tile (16 rows x kvalid cols) coalesced: per instruction,
        // 2 rows x 64 cols = two contiguous 256B segments.
        #pragma unroll
        for (int i = 0; i < 8; ++i) {
            int row  = 2 * i + half;      // 0..15
            int col4 = m * 4;             // 0..60
            if (col4 < kvalid) {
                v4f t = *(const v4f*)(L + (size_t)(r0 + row) * NV + kb + col4);
                *(v4f*)(&ldsA[row][col4]) = t;
            }
        }
        __syncthreads();

        // Prefetch next K-block of this tile's L rows (global_prefetch_b8).
        if (kb + KBLK < NV)
            __builtin_prefetch(L + (size_t)(r0 + m) * NV + kb + KBLK + half * 32, 0, 0);

        const int nsteps = kvalid >> 2;   // 16 or 4 WMMA K-steps
        for (int s = 0; s < nsteps; ++s) {
            const int ko = 4 * s + 2 * half;
            // A frag: lanes 0-15 -> K=k0,k0+1 ; lanes 16-31 -> K=k0+2,k0+3
            v2f a  = *(const v2f*)(&ldsA[m][ko]);
            // B frags: same K split, N = lane%16; Xt rows are contiguous in K
            v2f b0 = *(const v2f*)(Xt + (size_t)m        * NV + kb + ko);
            v2f b1 = *(const v2f*)(Xt + (size_t)(16 + m) * NV + kb + ko);
            // D = A x B + C   (v_wmma_f32_16x16x4_f32)
            c0 = __builtin_amdgcn_wmma_f32_16x16x4_f32(
                    false, a, false, b0, (short)0, c0, false, false);
            c1 = __builtin_amdgcn_wmma_f32_16x16x4_f32(
                    false, a, false, b1, (short)0, c1, false, false);
        }
        __syncthreads();
    }

    // Per-lane sum of squares over the 8 accumulator rows of each tile.
    float s0 = 0.0f, s1 = 0.0f;
    #pragma unroll
    for (int v = 0; v < 8; ++v) {
        s0 += c0[v] * c0[v];
        s1 += c1[v] * c1[v];
    }
    red[l]      = s0;   // column (l&15), M rows {half*8 .. half*8+7}
    red[32 + l] = s1;   // column 16+(l&15)
    __syncthreads();

    // Deterministic per-batch reduction: batch b owns columns 3b..3b+2.
    if (l < 8) {
        float t = 0.0f;
        #pragma unroll
        for (int q = 0; q < 3; ++q) {
            int c = 3 * l + q;
            if (c < 16) t += red[c]      + red[c + 16];
            else        t += red[c + 16] + red[c + 32];  // red[32+(c-16)] + red[48+(c-16)]
        }
        partial[tile * 8 + l] = t;
    }
}

// --- Kernel 3: fixed-order sum over 625 tile partials, sqrt, * NV.
__global__ void laploss_finalize(const float* __restrict__ partial,
                                 float* __restrict__ out) {
    int b = threadIdx.x;
    if (b < BATCH) {
        float s = 0.0f;
        for (int t = 0; t < TILES; ++t) s += partial[t * 8 + b];
        out[b] = sqrtf(s) * (float)NV;
    }
}

extern "C" void kernel_launch(void* const* d_in, const int* in_sizes, int n_in,
                              void* d_out, int out_size, void* d_ws, size_t ws_size,
                              hipStream_t stream) {
    const float* x = (const float*)d_in[0];   // (8, 10000, 3) f32
    const float* L = (const float*)d_in[1];   // (10000, 10000) f32
    const float* w = (const float*)d_in[2];   // scalar f32
    float* out     = (float*)d_out;           // (8,) f32

    float* Xt      = (float*)d_ws;            // 32*NV floats (1.28 MB)
    float* partial = Xt + (size_t)NCOLP * NV; // TILES*8 floats (20 KB)

    laploss_prep_xt<<<(NCOLP * NV + 255) / 256, 256, 0, stream>>>(x, w, Xt);
    laploss_gemm_tile<<<TILES, 32, 0, stream>>>(L, Xt, partial);
    laploss_finalize<<<1, 32, 0, stream>>>(partial, out);
}